// OutlierFeatureAggregator_62826781605922
// MI455X (gfx1250) — compile-verified
//
#include <hip/hip_runtime.h>
#include <math.h>

// Problem constants (match reference)
#define BB     16      // batch
#define NPTS   32768   // N points
#define DD     256     // point feature dim
#define EE     512     // text embed dim
#define HH     8       // heads
#define DH     64      // head dim
#define KSEL   512     // top-k
#define TT     16      // output tokens

typedef float v2f __attribute__((ext_vector_type(2)));
typedef float v8f __attribute__((ext_vector_type(8)));

// fp32 WMMA: D(16x16,f32) = A(16x4,f32) * B(4x16,f32) + C
__device__ __forceinline__ v8f wmma4(v2f a, v2f b, v8f c) {
  return __builtin_amdgcn_wmma_f32_16x16x4_f32(false, a, false, b, (short)0, c,
                                               false, false);
}

__device__ __forceinline__ unsigned int makeKey(float f) {
  unsigned int u = __float_as_uint(f);
  return u ^ ((u & 0x80000000u) ? 0xFFFFFFFFu : 0x80000000u); // ascending map
}

// ---------------------------------------------------------------------------
// Kernel 1: exact top-512 per batch. Radix-select threshold (4 byte passes),
// stable compaction (ties resolved to smallest index, matching jax top_k),
// then bitonic sort of the 512 winners (desc score, asc index).
// ---------------------------------------------------------------------------
__global__ __launch_bounds__(1024) void topk_kernel(
    const float* __restrict__ scores, int* __restrict__ idx_out) {
  const int b = blockIdx.x;
  const float* s = scores + (long long)b * NPTS;
  const int tid = threadIdx.x;

  __shared__ unsigned int hist[256];
  __shared__ unsigned int sc[1024];
  __shared__ unsigned long long sel[KSEL];
  __shared__ unsigned int sh_prefix, sh_remaining;
  __shared__ unsigned int sh_cntGt, sh_cntEq;

  if (tid == 0) { sh_prefix = 0u; sh_remaining = KSEL; }

  unsigned int prefMask = 0u;
  for (int shift = 24; shift >= 0; shift -= 8) {
    if (tid < 256) hist[tid] = 0u;
    __syncthreads();
    unsigned int pref = sh_prefix;
    for (int i = tid; i < NPTS; i += 1024) {
      unsigned int key = makeKey(s[i]);
      if ((key & prefMask) == pref)
        atomicAdd(&hist[(key >> shift) & 0xFFu], 1u);
    }
    __syncthreads();
    if (tid == 0) {
      unsigned int rem = sh_remaining, cum = 0u;
      int v = 255;
      for (; v > 0; --v) {
        if (cum + hist[v] >= rem) break;
        cum += hist[v];
      }
      sh_remaining = rem - cum;
      sh_prefix = pref | ((unsigned int)v << shift);
    }
    __syncthreads();
    prefMask |= 0xFFu << shift;
  }

  const unsigned int thr = sh_prefix;
  const unsigned int needEq = sh_remaining;      // #ties to take
  const unsigned int baseEq = KSEL - needEq;     // = total strictly-greater
  if (tid == 0) { sh_cntGt = 0u; sh_cntEq = 0u; }
  __syncthreads();

  // stable compaction: chunks of 1024, Hillis-Steele block scan
  for (int c0 = 0; c0 < NPTS; c0 += 1024) {
    const int i = c0 + tid;
    const unsigned int key = makeKey(s[i]);
    const unsigned int pGt = (key > thr) ? 1u : 0u;
    const unsigned int pEq = (key == thr) ? 1u : 0u;
    sc[tid] = pGt | (pEq << 16);
    __syncthreads();
    for (int off = 1; off < 1024; off <<= 1) {
      unsigned int v = (tid >= off) ? sc[tid - off] : 0u;
      __syncthreads();
      sc[tid] += v;
      __syncthreads();
    }
    const unsigned int incl = sc[tid];
    const unsigned int exGt = (incl & 0xFFFFu) - pGt;
    const unsigned int exEq = (incl >> 16) - pEq;
    const unsigned int cGt = sh_cntGt, cEq = sh_cntEq;
    const unsigned long long pk =
        ((unsigned long long)key << 32) | (unsigned int)(~(unsigned int)i);
    if (pGt) sel[cGt + exGt] = pk;
    if (pEq) {
      unsigned int pos = cEq + exEq;
      if (pos < needEq) sel[baseEq + pos] = pk;
    }
    __syncthreads();
    if (tid == 1023) {
      sh_cntGt += (incl & 0xFFFFu);
      sh_cntEq += (incl >> 16);
    }
    __syncthreads();
  }

  // bitonic sort: descending by key, ascending index (via ~idx in low bits)
  for (int ksz = 2; ksz <= KSEL; ksz <<= 1) {
    for (int j = ksz >> 1; j > 0; j >>= 1) {
      if (tid < KSEL) {
        int ixj = tid ^ j;
        if (ixj > tid) {
          unsigned long long a = sel[tid], c = sel[ixj];
          bool descRegion = ((tid & ksz) == 0);
          if (descRegion ? (a < c) : (a > c)) { sel[tid] = c; sel[ixj] = a; }
        }
      }
      __syncthreads();
    }
  }
  if (tid < KSEL)
    idx_out[(long long)b * KSEL + tid] =
        (int)(~(unsigned int)(sel[tid] & 0xFFFFFFFFull));
}

// ---------------------------------------------------------------------------
// Kernel 2: generic C[M,N] = A[M,KD] @ W[N,KD]^T + bias, optional row gather
// on A. 64x64 block tile, BK=32, 8 waves x two 16x16 WMMA accumulators.
// ---------------------------------------------------------------------------
#define BM 64
#define BN 64
#define BK 32

__global__ __launch_bounds__(256) void gemm_nt_kernel(
    const float* __restrict__ Abase, long long strideA, int lda,
    const int* __restrict__ gatherBase, int gatherStride,
    const float* __restrict__ W, int ldw, const float* __restrict__ bias,
    float* __restrict__ Cbase, long long strideC, int ldc,
    int M, int N, int KD) {
  const int b = blockIdx.z;
  const float* A = Abase + (long long)b * strideA;
  float* C = Cbase + (long long)b * strideC;
  const int* gidx =
      gatherBase ? (gatherBase + (long long)b * gatherStride) : nullptr;

  const int m0 = blockIdx.y * BM;
  const int n0 = blockIdx.x * BN;

  __shared__ float As[BK][BM + 1];
  __shared__ float Ws[BK][BN + 1];

  const int tid = threadIdx.x;
  const int lane = tid & 31, wave = tid >> 5;
  const int lm = lane & 15, half = lane >> 4;
  const int cc = tid & 31;  // column within BK
  const int rr = tid >> 5;  // base row (0..7)

  const int tile0 = wave * 2;      // even tile id -> same tm for both tiles
  const int tm = tile0 >> 2;
  const int tn0 = tile0 & 3, tn1 = tn0 + 1;

  v8f acc0 = {}; v8f acc1 = {};

  for (int k0 = 0; k0 < KD; k0 += BK) {
    for (int r = rr; r < BM; r += 8) {
      int m = m0 + r;
      float v = 0.f;
      if (m < M) {
        int row = gidx ? gidx[m] : m;
        v = A[(long long)row * lda + (k0 + cc)];
      }
      As[cc][r] = v;
    }
    for (int r = rr; r < BN; r += 8) {
      int n = n0 + r;
      Ws[cc][r] = (n < N) ? W[(long long)n * ldw + (k0 + cc)] : 0.f;
    }
    __syncthreads();

#pragma unroll
    for (int k4 = 0; k4 < BK; k4 += 4) {
      v2f a, w0, w1;
      a.x = As[k4 + 0 + 2 * half][tm * 16 + lm];
      a.y = As[k4 + 1 + 2 * half][tm * 16 + lm];
      w0.x = Ws[k4 + 0 + 2 * half][tn0 * 16 + lm];
      w0.y = Ws[k4 + 1 + 2 * half][tn0 * 16 + lm];
      w1.x = Ws[k4 + 0 + 2 * half][tn1 * 16 + lm];
      w1.y = Ws[k4 + 1 + 2 * half][tn1 * 16 + lm];
      acc0 = wmma4(a, w0, acc0);
      acc1 = wmma4(a, w1, acc1);
    }
    __syncthreads();
  }

  for (int t = 0; t < 2; ++t) {
    const int tn = (t == 0) ? tn0 : tn1;
    const v8f acc = (t == 0) ? acc0 : acc1;
    const int nn = n0 + tn * 16 + lm;
#pragma unroll
    for (int j = 0; j < 8; ++j) {
      int m = m0 + tm * 16 + j + 8 * half;
      if (m < M && nn < N)
        C[(long long)m * ldc + nn] = acc[j] + (bias ? bias[nn] : 0.f);
    }
  }
}

// ---------------------------------------------------------------------------
// Kernel 3: attention per (head, batch). 16 queries x 512 keys, WMMA both
// GEMMs, serial-row softmax (16 rows).
// ---------------------------------------------------------------------------
__global__ __launch_bounds__(256) void attn_kernel(
    const float* __restrict__ qbuf,  // [B,T,E]
    const float* __restrict__ kmat,  // [B,K,E]
    const float* __restrict__ vmat,  // [B,K,E]
    float* __restrict__ obuf) {      // [B,T,E]
  const int h = blockIdx.x;
  const int b = blockIdx.y;
  __shared__ float qs[TT][DH];
  __shared__ float ls[TT][KSEL];
  const int tid = threadIdx.x, lane = tid & 31, wave = tid >> 5;
  const int lm = lane & 15, half = lane >> 4;

  for (int i = tid; i < TT * DH; i += 256)
    qs[i / DH][i % DH] = qbuf[((long long)b * TT + i / DH) * EE + h * DH + i % DH];
  __syncthreads();

  const float* kb = kmat + (long long)b * KSEL * EE + h * DH;
  for (int t = wave; t < KSEL / 16; t += 8) {  // 32 key tiles, 4 per wave
    v8f acc = {};
#pragma unroll
    for (int k4 = 0; k4 < DH; k4 += 4) {
      v2f a, w;
      a.x = qs[lm][k4 + 0 + 2 * half];
      a.y = qs[lm][k4 + 1 + 2 * half];
      w.x = kb[(long long)(t * 16 + lm) * EE + (k4 + 0 + 2 * half)];
      w.y = kb[(long long)(t * 16 + lm) * EE + (k4 + 1 + 2 * half)];
      acc = wmma4(a, w, acc);
    }
#pragma unroll
    for (int j = 0; j < 8; ++j)
      ls[j + 8 * half][t * 16 + lm] = acc[j] * 0.125f;  // 1/sqrt(64)
  }
  __syncthreads();

  if (tid < TT) {
    float mx = -3.4e38f;
    for (int i = 0; i < KSEL; ++i) mx = fmaxf(mx, ls[tid][i]);
    float sum = 0.f;
    for (int i = 0; i < KSEL; ++i) {
      float e = expf(ls[tid][i] - mx);
      ls[tid][i] = e;
      sum += e;
    }
    float inv = 1.f / sum;
    for (int i = 0; i < KSEL; ++i) ls[tid][i] *= inv;
  }
  __syncthreads();

  const float* vb = vmat + (long long)b * KSEL * EE + h * DH;
  if (wave < 4) {          // 4 dh-tiles of 16; wave-uniform branch (EXEC full)
    const int t = wave;
    v8f acc = {};
    for (int k4 = 0; k4 < KSEL; k4 += 4) {
      v2f a, w;
      a.x = ls[lm][k4 + 0 + 2 * half];
      a.y = ls[lm][k4 + 1 + 2 * half];
      w.x = vb[(long long)(k4 + 0 + 2 * half) * EE + t * 16 + lm];
      w.y = vb[(long long)(k4 + 1 + 2 * half) * EE + t * 16 + lm];
      acc = wmma4(a, w, acc);
    }
#pragma unroll
    for (int j = 0; j < 8; ++j)
      obuf[((long long)b * TT + (j + 8 * half)) * EE + h * DH + t * 16 + lm] =
          acc[j];
  }
}

// ---------------------------------------------------------------------------
// Kernel 4: out_proj + residual + LayerNorm (per batch, 16 rows)
// ---------------------------------------------------------------------------
__global__ __launch_bounds__(256) void outproj_ln_kernel(
    const float* __restrict__ obuf, const float* __restrict__ xbuf,
    const float* __restrict__ Wo, const float* __restrict__ bo,
    const float* __restrict__ ln_g, const float* __restrict__ ln_b,
    float* __restrict__ ybuf) {
  const int b = blockIdx.x;
  __shared__ float os[TT][EE];  // 32 KB
  const int tid = threadIdx.x, lane = tid & 31, wave = tid >> 5;
  const int lm = lane & 15, half = lane >> 4;

  for (int i = tid; i < TT * EE; i += 256)
    os[i / EE][i % EE] = obuf[(long long)b * TT * EE + i];
  __syncthreads();

  v8f acc[4];
  for (int tt = 0; tt < 4; ++tt) {
    const int t = wave * 4 + tt;  // 32 n-tiles of 16
    v8f a8 = {};
    for (int k4 = 0; k4 < EE; k4 += 4) {
      v2f a, w;
      a.x = os[lm][k4 + 0 + 2 * half];
      a.y = os[lm][k4 + 1 + 2 * half];
      w.x = Wo[(long long)(t * 16 + lm) * EE + (k4 + 0 + 2 * half)];
      w.y = Wo[(long long)(t * 16 + lm) * EE + (k4 + 1 + 2 * half)];
      a8 = wmma4(a, w, a8);
    }
    acc[tt] = a8;
  }
  __syncthreads();  // all reads of os complete before in-place overwrite

  for (int tt = 0; tt < 4; ++tt) {
    const int t = wave * 4 + tt;
    const int nn = t * 16 + lm;
#pragma unroll
    for (int j = 0; j < 8; ++j) {
      const int r = j + 8 * half;
      os[r][nn] = acc[tt][j] + bo[nn] +
                  xbuf[((long long)b * KSEL + r) * EE + nn];  // residual
    }
  }
  __syncthreads();

  if (tid < TT) {
    float mu = 0.f;
    for (int i = 0; i < EE; ++i) mu += os[tid][i];
    mu *= (1.f / EE);
    float var = 0.f;
    for (int i = 0; i < EE; ++i) {
      float d = os[tid][i] - mu;
      var += d * d;
    }
    var *= (1.f / EE);
    const float inv = rsqrtf(var + 1e-5f);
    for (int i = 0; i < EE; ++i)
      ybuf[(long long)b * TT * EE + tid * EE + i] =
          (os[tid][i] - mu) * inv * ln_g[i] + ln_b[i];
  }
}

// ---------------------------------------------------------------------------
// Kernel 5: FFN (linear + exact GELU) + residual -> final output
// ---------------------------------------------------------------------------
__global__ __launch_bounds__(256) void ffn_kernel(
    const float* __restrict__ ybuf, const float* __restrict__ Wf,
    const float* __restrict__ bf, float* __restrict__ outp) {
  const int b = blockIdx.x;
  __shared__ float ys[TT][EE];
  const int tid = threadIdx.x, lane = tid & 31, wave = tid >> 5;
  const int lm = lane & 15, half = lane >> 4;

  for (int i = tid; i < TT * EE; i += 256)
    ys[i / EE][i % EE] = ybuf[(long long)b * TT * EE + i];
  __syncthreads();

  for (int tt = 0; tt < 4; ++tt) {
    const int t = wave * 4 + tt;
    v8f a8 = {};
    for (int k4 = 0; k4 < EE; k4 += 4) {
      v2f a, w;
      a.x = ys[lm][k4 + 0 + 2 * half];
      a.y = ys[lm][k4 + 1 + 2 * half];
      w.x = Wf[(long long)(t * 16 + lm) * EE + (k4 + 0 + 2 * half)];
      w.y = Wf[(long long)(t * 16 + lm) * EE + (k4 + 1 + 2 * half)];
      a8 = wmma4(a, w, a8);
    }
    const int nn = t * 16 + lm;
#pragma unroll
    for (int j = 0; j < 8; ++j) {
      const int r = j + 8 * half;
      float g = a8[j] + bf[nn];
      float gelu = 0.5f * g * (1.f + erff(g * 0.70710678118654752f));
      outp[(long long)b * TT * EE + (long long)r * EE + nn] = ys[r][nn] + gelu;
    }
  }
}

// ---------------------------------------------------------------------------
extern "C" void kernel_launch(void* const* d_in, const int* in_sizes, int n_in,
                              void* d_out, int out_size, void* d_ws,
                              size_t ws_size, hipStream_t stream) {
  const float* feats   = (const float*)d_in[0];   // [B,N,D]
  const float* scores  = (const float*)d_in[1];   // [B,N,1]
  const float* proj_w  = (const float*)d_in[2];   // [E,D]
  const float* proj_b  = (const float*)d_in[3];
  const float* ipw     = (const float*)d_in[4];   // [3E,E]
  const float* ipb     = (const float*)d_in[5];
  const float* opw     = (const float*)d_in[6];   // [E,E]
  const float* opb     = (const float*)d_in[7];
  const float* ln_g    = (const float*)d_in[8];
  const float* ln_b    = (const float*)d_in[9];
  const float* ffn_w   = (const float*)d_in[10];  // [E,E]
  const float* ffn_b   = (const float*)d_in[11];
  float* out = (float*)d_out;

  // workspace carve (256B aligned)
  char* ws = (char*)d_ws;
  auto carve = [&](size_t bytes) {
    char* p = ws;
    ws += (bytes + 255) & ~(size_t)255;
    return p;
  };
  int* idx_sel = (int*)carve(sizeof(int) * BB * KSEL);
  float* xbuf  = (float*)carve(sizeof(float) * (size_t)BB * KSEL * EE);
  float* kmat  = (float*)carve(sizeof(float) * (size_t)BB * KSEL * EE);
  float* vmat  = (float*)carve(sizeof(float) * (size_t)BB * KSEL * EE);
  float* qbuf  = (float*)carve(sizeof(float) * (size_t)BB * TT * EE);
  float* obuf  = (float*)carve(sizeof(float) * (size_t)BB * TT * EE);
  float* ybuf  = (float*)carve(sizeof(float) * (size_t)BB * TT * EE);
  (void)in_sizes; (void)n_in; (void)out_size; (void)ws_size;

  topk_kernel<<<BB, 1024, 0, stream>>>(scores, idx_sel);

  dim3 blk(256);
  // x = gather(feats) @ proj_w^T + proj_b   [512x512x256 per batch]
  gemm_nt_kernel<<<dim3(EE / 64, KSEL / 64, BB), blk, 0, stream>>>(
      feats, (long long)NPTS * DD, DD, idx_sel, KSEL, proj_w, DD, proj_b, xbuf,
      (long long)KSEL * EE, EE, KSEL, EE, DD);
  // k = x @ Wk^T + bk
  gemm_nt_kernel<<<dim3(EE / 64, KSEL / 64, BB), blk, 0, stream>>>(
      xbuf, (long long)KSEL * EE, EE, nullptr, 0, ipw + (size_t)EE * EE, EE,
      ipb + EE, kmat, (long long)KSEL * EE, EE, KSEL, EE, EE);
  // v = x @ Wv^T + bv
  gemm_nt_kernel<<<dim3(EE / 64, KSEL / 64, BB), blk, 0, stream>>>(
      xbuf, (long long)KSEL * EE, EE, nullptr, 0, ipw + (size_t)2 * EE * EE, EE,
      ipb + 2 * EE, vmat, (long long)KSEL * EE, EE, KSEL, EE, EE);
  // q: only the first 16 rows are ever needed by the output
  gemm_nt_kernel<<<dim3(EE / 64, 1, BB), blk, 0, stream>>>(
      xbuf, (long long)KSEL * EE, EE, nullptr, 0, ipw, EE, ipb, qbuf,
      (long long)TT * EE, EE, TT, EE, EE);

  attn_kernel<<<dim3(HH, BB), blk, 0, stream>>>(qbuf, kmat, vmat, obuf);
  outproj_ln_kernel<<<BB, blk, 0, stream>>>(obuf, xbuf, opw, opb, ln_g, ln_b,
                                            ybuf);
  ffn_kernel<<<BB, blk, 0, stream>>>(ybuf, ffn_w, ffn_b, out);
}